// PointLinkTrainer_10831907521159
// MI455X (gfx1250) — compile-verified
//
#include <hip/hip_runtime.h>

#define G      14
#define NCH    51                 // 3 + 20 + 2*14
#define CELLS  (G * G)            // 196
#define NFLOAT (CELLS * 4 * NCH)  // 39984 floats per image
#define NVEC   (NFLOAT / 4)       // 9996 float4 per image
#define BLK    256

typedef __attribute__((ext_vector_type(2))) float v2f;
typedef __attribute__((ext_vector_type(4))) float v4f;
typedef __attribute__((ext_vector_type(8))) float v8f;

__global__ __launch_bounds__(64) void pl_zero(float* p) {
  if (threadIdx.x == 0) p[0] = 0.0f;
}

__global__ __launch_bounds__(BLK) void pl_loss(
    const float* __restrict__ outp,
    const float* __restrict__ bboxes,
    const int*   __restrict__ labels,
    float*       __restrict__ loss) {
  __shared__ float gt[CELLS * NCH];  // dense per-grid-cell target (cell-major, channels contiguous)
  __shared__ float wlut[NCH];        // per-channel loss weights
  __shared__ float wred[16];         // cross-wave partials (2 per wave)

  const int tid = threadIdx.x;
  const int n   = blockIdx.x;

  // ---- zero targets + build weight LUT --------------------------------
  for (int i = tid; i < CELLS * NCH; i += BLK) gt[i] = 0.0f;
  if (tid < NCH) {
    float w = 1.0f;                       // conf (0), x (21), y (22)
    if (tid >= 1 && tid <= 20) w = 1.0f / 20.0f;   // class mean
    else if (tid >= 23)        w = 1.0f / 14.0f;   // link means
    wlut[tid] = w;
  }
  __syncthreads();

  // ---- scatter ground truth: 8 boxes x 4 corner points ----------------
  if (tid < 32) {
    const int m = tid >> 2;
    const int k = tid & 3;
    const float* bb = bboxes + ((size_t)n * 8 + m) * 4;
    const float s = (float)G / 448.0f;
    float gx = bb[0] * s, gy = bb[1] * s, gw = bb[2] * s, gh = bb[3] * s;
    float px = (k & 1) ? (gx + gw) : gx;
    float py = (k & 2) ? (gy + gh) : gy;
    float fpx = floorf(px), fpy = floorf(py);
    int ix = min(G - 1, max(0, (int)fpx));
    int iy = min(G - 1, max(0, (int)fpy));
    float fx = px - fpx;
    float fy = py - fpy;
    int cx = min(G - 1, max(0, (int)floorf(gx + 0.5f * gw)));
    int cy = min(G - 1, max(0, (int)floorf(gy + 0.5f * gh)));
    int lab = labels[(size_t)n * 8 + m];
    float* cell = gt + (ix * G + iy) * NCH;
    cell[0]        = 1.0f;   // conf target (also the mask)
    cell[1 + lab]  = 1.0f;   // class one-hot
    cell[21]       = fx;     // coord targets
    cell[22]       = fy;
    cell[23 + cx]  = 1.0f;   // link-x one-hot
    cell[37 + cy]  = 1.0f;   // link-y one-hot
  }
  __syncthreads();

  // ---- stream this image's predictions (bandwidth-bound) --------------
  const v4f* src = (const v4f*)(outp + (size_t)n * NFLOAT);
  float acc = 0.0f;
  for (int i = tid; i < NVEC; i += BLK) {
    if (i + 2 * BLK < NVEC)
      __builtin_prefetch((const void*)(src + i + 2 * BLK), 0, 1);
    v4f v = __builtin_nontemporal_load(src + i);  // no reuse: keep L2 clean
    const int f = i * 4;
#pragma unroll
    for (int e = 0; e < 4; ++e) {
      int idx = f + e;
      int c4  = idx / NCH;             // cell index among 196*4
      int ch  = idx - c4 * NCH;        // channel 0..50
      const float* tgt = gt + (c4 >> 2) * NCH;  // b-dim shares target
      float conf = tgt[0];
      float vel  = v[e];
      float d, w;
      if (conf > 0.0f) { d = vel - tgt[ch]; w = wlut[ch]; }          // loss_pos
      else             { d = vel;           w = (ch == 0) ? 1.0f : 0.0f; } // loss_neg
      acc = fmaf(w * d, d, acc);
    }
  }

  // ---- exact f32 wave reduction on the matrix core --------------------
  // A (16x4 f32): lane m holds (M=m,K=0), lane m+16 holds (M=m,K=2); K=1,3 zero.
  // B = ones  =>  D[m,n] = acc[m] + acc[m+16] for every n.
  // Sum of a lane's 8 D VGPRs = sum over its half of M => lanes 0 and 16
  // together hold the full 32-lane total.
  v2f a; a[0] = acc;  a[1] = 0.0f;
  v2f b; b[0] = 1.0f; b[1] = 1.0f;
  v8f c = {};
  v8f dmat = __builtin_amdgcn_wmma_f32_16x16x4_f32(
      /*neg_a=*/false, a, /*neg_b=*/false, b,
      /*c_mod=*/(short)0, c, /*reuse_a=*/false, /*reuse_b=*/false);
  float half_sum = dmat[0] + dmat[1] + dmat[2] + dmat[3] +
                   dmat[4] + dmat[5] + dmat[6] + dmat[7];

  const int lane = tid & 31;
  const int wave = tid >> 5;
  if (lane == 0)  wred[wave * 2 + 0] = half_sum;
  if (lane == 16) wred[wave * 2 + 1] = half_sum;
  __syncthreads();
  if (tid == 0) {
    float stot = 0.0f;
#pragma unroll
    for (int j = 0; j < 16; ++j) stot += wred[j];
    atomicAdd(loss, stot);
  }
}

extern "C" void kernel_launch(void* const* d_in, const int* in_sizes, int n_in,
                              void* d_out, int out_size, void* d_ws, size_t ws_size,
                              hipStream_t stream) {
  const float* outp   = (const float*)d_in[0];
  const float* bboxes = (const float*)d_in[1];
  const int*   labels = (const int*)d_in[2];
  float* loss = (float*)d_out;

  const int N = in_sizes[0] / NFLOAT;  // 4096

  pl_zero<<<1, 64, 0, stream>>>(loss);
  pl_loss<<<N, BLK, 0, stream>>>(outp, bboxes, labels, loss);
}